// NetGCN_54683523612721
// MI455X (gfx1250) — compile-verified
//
#include <hip/hip_runtime.h>
#include <hip/hip_bf16.h>
#include <hip/hip_fp16.h>

#define N_NODES  50000
#define N_EDGES  800000
#define NF       128
#define N_GRAPHS 512

#define ROWS_PER_BLOCK 64          // 4 row-tiles of 16, per block

typedef __attribute__((ext_vector_type(2)))  float v2f;
typedef __attribute__((ext_vector_type(8)))  float v8f;

// ---------------------------------------------------------------- zero fill
__global__ __launch_bounds__(256) void zerof(float* __restrict__ p, int n) {
    int i = blockIdx.x * 256 + threadIdx.x;
    if (i < n) p[i] = 0.0f;
}

// ---------------------------------------------------------------- GEMM (WMMA)
// H[0:nrows,0:128] = act(X) @ W      X:[nrows,128] W:[128,128] row-major
// Block: 256 threads = 8 waves; block computes 64 rows x 128 cols.
// Wave w owns columns [16w,16w+16) and 4 row-tiles; B fragment loaded once
// per K-step and reused by 4 chained v_wmma_f32_16x16x4_f32 ops.
__global__ __launch_bounds__(256) void gemm_wmma(const float* __restrict__ X,
                                                 const float* __restrict__ W,
                                                 float* __restrict__ H,
                                                 int nrows, int relu_in) {
    __shared__ float sX[ROWS_PER_BLOCK * NF];           // 32 KB
    const int row0 = blockIdx.x * ROWS_PER_BLOCK;

    // cooperative tile load, fused ReLU, zero-pad tail rows
    for (int idx = threadIdx.x; idx < ROWS_PER_BLOCK * NF; idx += 256) {
        int r = idx >> 7, c = idx & (NF - 1);
        int row = row0 + r;
        float v = (row < nrows) ? X[(size_t)row * NF + c] : 0.0f;
        if (relu_in) v = fmaxf(v, 0.0f);
        sX[idx] = v;
    }
    __syncthreads();

    const int wave  = threadIdx.x >> 5;
    const int lane  = threadIdx.x & 31;
    const int col0  = wave * 16;
    const int lhalf = lane >> 4;      // 0: lanes 0-15, 1: lanes 16-31
    const int lmod  = lane & 15;

    v8f acc0 = {}, acc1 = {}, acc2 = {}, acc3 = {};

    // A layout (ISA 7.12.2, 32-bit A 16x4): lane<16 -> K={k,k+1}; lane>=16 -> K={k+2,k+3}
    // B layout (row striped across lanes): lane-half selects K pair, lmod = column.
    for (int k = 0; k < NF; k += 4) {
        const int ka = k + lhalf * 2;
        v2f b;
        b.x = W[(size_t)ka       * NF + col0 + lmod];
        b.y = W[(size_t)(ka + 1) * NF + col0 + lmod];

        const float* s = sX + lmod * NF + ka;
        v2f a0, a1, a2, a3;
        a0.x = s[0 * 16 * NF]; a0.y = s[0 * 16 * NF + 1];
        a1.x = s[1 * 16 * NF]; a1.y = s[1 * 16 * NF + 1];
        a2.x = s[2 * 16 * NF]; a2.y = s[2 * 16 * NF + 1];
        a3.x = s[3 * 16 * NF]; a3.y = s[3 * 16 * NF + 1];

        acc0 = __builtin_amdgcn_wmma_f32_16x16x4_f32(false, a0, false, b, (short)0, acc0, false, false);
        acc1 = __builtin_amdgcn_wmma_f32_16x16x4_f32(false, a1, false, b, (short)0, acc1, false, false);
        acc2 = __builtin_amdgcn_wmma_f32_16x16x4_f32(false, a2, false, b, (short)0, acc2, false, false);
        acc3 = __builtin_amdgcn_wmma_f32_16x16x4_f32(false, a3, false, b, (short)0, acc3, false, false);
    }

    // D layout: VGPR i -> M = i + 8*lane_half, N = lmod
    const int mbase = lhalf * 8;
    #pragma unroll
    for (int i = 0; i < 8; ++i) {
        int r0 = row0 + 0 * 16 + mbase + i;
        int r1 = row0 + 1 * 16 + mbase + i;
        int r2 = row0 + 2 * 16 + mbase + i;
        int r3 = row0 + 3 * 16 + mbase + i;
        if (r0 < nrows) H[(size_t)r0 * NF + col0 + lmod] = acc0[i];
        if (r1 < nrows) H[(size_t)r1 * NF + col0 + lmod] = acc1[i];
        if (r2 < nrows) H[(size_t)r2 * NF + col0 + lmod] = acc2[i];
        if (r3 < nrows) H[(size_t)r3 * NF + col0 + lmod] = acc3[i];
    }
}

// ---------------------------------------------------------------- edge scatter
// agg[dst[e]] += H[src[e]]  ; one thread per (edge, 4 floats). Node features
// (25.6 MB) are L2-resident on MI455X (192 MB L2), so gathers and the f32
// atomics stay in L2; HBM sees only first-touch traffic.
__global__ __launch_bounds__(256) void scatter_add(const float* __restrict__ H,
                                                   const long long* __restrict__ src,
                                                   const long long* __restrict__ dst,
                                                   float* __restrict__ agg) {
    long long t = (long long)blockIdx.x * 256 + threadIdx.x;
    long long e = t >> 5;                 // wave-uniform edge id
    if (e >= N_EDGES) return;
    int c = (int)(t & 31) << 2;
    long long s = src[e];
    long long d = dst[e];
    const float4 v = *(const float4*)(H + s * NF + c);
    float* out = agg + d * NF + c;
    atomicAdd(out + 0, v.x);
    atomicAdd(out + 1, v.y);
    atomicAdd(out + 2, v.z);
    atomicAdd(out + 3, v.w);
}

// ---------------------------------------------------------------- graph pool
__global__ __launch_bounds__(256) void pool_sum(const float* __restrict__ A,
                                                const long long* __restrict__ batch,
                                                float* __restrict__ sums,
                                                float* __restrict__ counts) {
    long long t = (long long)blockIdx.x * 256 + threadIdx.x;
    long long node = t >> 5;
    if (node >= N_NODES) return;
    int c = (int)(t & 31) << 2;
    long long g = batch[node];
    const float4 v = *(const float4*)(A + node * NF + c);
    float* out = sums + g * NF + c;
    atomicAdd(out + 0, v.x);
    atomicAdd(out + 1, v.y);
    atomicAdd(out + 2, v.z);
    atomicAdd(out + 3, v.w);
    if (c == 0) atomicAdd(&counts[g], 1.0f);
}

// ---------------------------------------------------------------- head
// out[g] = sigmoid( (sums[g] . Wfc) / max(count[g],1) )   one wave per graph
__global__ __launch_bounds__(256) void head(const float* __restrict__ sums,
                                            const float* __restrict__ counts,
                                            const float* __restrict__ Wfc,
                                            float* __restrict__ out) {
    int g    = (blockIdx.x * 256 + threadIdx.x) >> 5;
    int lane = threadIdx.x & 31;
    if (g >= N_GRAPHS) return;
    float partial = 0.0f;
    #pragma unroll
    for (int c = lane; c < NF; c += 32)
        partial += sums[(size_t)g * NF + c] * Wfc[c];
    #pragma unroll
    for (int off = 16; off > 0; off >>= 1)
        partial += __shfl_down(partial, off, 32);
    if (lane == 0) {
        float cnt = fmaxf(counts[g], 1.0f);
        float z = partial / cnt;
        out[g] = 1.0f / (1.0f + __expf(-z));
    }
}

// ---------------------------------------------------------------- launcher
extern "C" void kernel_launch(void* const* d_in, const int* in_sizes, int n_in,
                              void* d_out, int out_size, void* d_ws, size_t ws_size,
                              hipStream_t stream) {
    const float*     x     = (const float*)d_in[0];
    const long long* edge  = (const long long*)d_in[1];   // int64 [2, N_EDGES]
    const long long* batch = (const long long*)d_in[2];   // int64 [N_NODES]
    const float*     W1    = (const float*)d_in[3];
    const float*     W2    = (const float*)d_in[4];
    const float*     Wfc   = (const float*)d_in[5];
    float*           out   = (float*)d_out;

    const long long* src = edge;
    const long long* dst = edge + N_EDGES;

    float* bufA   = (float*)d_ws;                         // 25.6 MB
    float* bufB   = bufA + (size_t)N_NODES * NF;          // 25.6 MB
    float* sums   = bufB + (size_t)N_NODES * NF;          // 256 KB
    float* counts = sums + (size_t)N_GRAPHS * NF;         // 2 KB

    const int NH = N_NODES * NF;
    dim3 blk(256);
    const int gemm_blocks    = (N_NODES + ROWS_PER_BLOCK - 1) / ROWS_PER_BLOCK; // 782
    const int scatter_blocks = (N_EDGES * 32) / 256;      // 100000
    const int pool_blocks    = (N_NODES * 32 + 255) / 256;

    // zero agg1 + pooled sums/counts (sums & counts contiguous)
    zerof<<<(NH + 255) / 256, blk, 0, stream>>>(bufB, NH);
    zerof<<<(N_GRAPHS * NF + N_GRAPHS + 255) / 256, blk, 0, stream>>>(
        sums, N_GRAPHS * NF + N_GRAPHS);

    // conv1: h1 = x @ W1 ; scatter into agg1
    gemm_wmma<<<gemm_blocks, blk, 0, stream>>>(x, W1, bufA, N_NODES, 0);
    scatter_add<<<scatter_blocks, blk, 0, stream>>>(bufA, src, dst, bufB);

    // conv2: h2 = relu(agg1) @ W2 (ReLU fused on GEMM input); scatter into agg2
    gemm_wmma<<<gemm_blocks, blk, 0, stream>>>(bufB, W2, bufA, N_NODES, 1);
    zerof<<<(NH + 255) / 256, blk, 0, stream>>>(bufB, NH);
    scatter_add<<<scatter_blocks, blk, 0, stream>>>(bufA, src, dst, bufB);

    // mean pool + fc + sigmoid
    pool_sum<<<pool_blocks, blk, 0, stream>>>(bufB, batch, sums, counts);
    head<<<(N_GRAPHS * 32 + 255) / 256, blk, 0, stream>>>(sums, counts, Wfc, out);
}